// SoftMSMLoss_8589934592318
// MI455X (gfx1250) — compile-verified
//
#include <hip/hip_runtime.h>
#include <cstdint>

#define N    1024
#define R    4
#define T    (N / R)
#define BIGF 1e30f

// trans(a,b) = 1 + (1-gate(a,b)) * softmin2(a^2, b^2), gamma=1, aa = a*a hoisted
__device__ __forceinline__ float transf2(float a, float aa, float b) {
    float u  = a * b;
    float g  = 0.5f * (1.0f - u * rsqrtf(__fmaf_rn(u, u, 1e-9f)));
    float bb = b * b;
    float mn = fminf(aa, bb);
    float mx = fmaxf(aa, bb);
    float sm = mn - __logf(1.0f + __expf(mn - mx));   // stable softmin2
    return __fmaf_rn(1.0f - g, sm, 1.0f);
}

__device__ __forceinline__ float transf(float a, float b) {
    return transf2(a, a * a, b);
}

__device__ __forceinline__ float softmin3f(float d1, float d2, float d3) {
    float mn = fminf(d1, fminf(d2, d3));
    float e  = __expf(mn - d1) + __expf(mn - d2) + __expf(mn - d3);
    return mn - __logf(e);
}

// One workgroup per batch element; thread t owns rows i0..i0+3 (i0 = 4t).
// Anti-diagonal wavefront, one barrier per diagonal. For rows k>0 the up/diag
// predecessors are this thread's own registers (cur[k-1] / oldv[k-1]); only
// row 0 reads the neighbor's bottom row from an LDS ping-pong buffer, and only
// the bottom row is published.
__global__ __launch_bounds__(T, 1) void soft_msm_kernel(
        const float* __restrict__ x, const float* __restrict__ y,
        float* __restrict__ ws) {
    __shared__ __align__(16) float  xs[N];
    __shared__ __align__(16) float  ysa[N];
    __shared__ __align__(16) float2 ydy[N];     // {y[j], y[j]-y[j-1]}
    __shared__ float buf[2][T];

    const int t  = threadIdx.x;
    const int b  = blockIdx.x;
    const int i0 = R * t;

    // --- CDNA5 async global->LDS staging (ASYNCcnt-tracked, 16B/thread/array) ---
    {
        const float* gx = x + (size_t)b * N + i0;
        const float* gy = y + (size_t)b * N + i0;
        unsigned lx = (unsigned)(uintptr_t)(&xs[i0]);
        unsigned ly = (unsigned)(uintptr_t)(&ysa[i0]);
        asm volatile("global_load_async_to_lds_b128 %0, %1, off" :: "v"(lx), "v"(gx) : "memory");
        asm volatile("global_load_async_to_lds_b128 %0, %1, off" :: "v"(ly), "v"(gy) : "memory");
        asm volatile("s_wait_asynccnt 0" ::: "memory");
    }
    __syncthreads();

    #pragma unroll
    for (int k = 0; k < R; ++k) {
        int i = i0 + k;
        float yi  = ysa[i];
        float dyi = (i > 0) ? (yi - ysa[i - 1]) : 0.0f;
        ydy[i] = make_float2(yi, dyi);
    }

    float xv[R], dxv[R], dx2[R];
    {
        float xm1 = (t > 0) ? xs[i0 - 1] : 0.0f;
        #pragma unroll
        for (int k = 0; k < R; ++k) {
            xv[k]  = xs[i0 + k];
            float p = (k == 0) ? xm1 : xv[k - 1];
            dxv[k] = xv[k] - p;
            dx2[k] = dxv[k] * dxv[k];
        }
    }
    __syncthreads();

    float cur[R], oldv[R], yv[R], dyv[R];
    #pragma unroll
    for (int k = 0; k < R; ++k) { cur[k] = BIGF; oldv[k] = BIGF; yv[k] = 0.0f; dyv[k] = 0.0f; }
    float dgN = BIGF;                        // neighbor bottom-row value, 2 steps back
    const int tm1 = (t > 0) ? (t - 1) : 0;

    for (int d = 0; d < 2 * N - 1; ++d) {
        const int j0 = d - i0;               // row 0's column this step
        if (j0 >= 0 && j0 <= (N - 1) + (R - 1)) {
            // y register pipeline: row k uses what row k-1 used last step
            #pragma unroll
            for (int k = R - 1; k >= 1; --k) { yv[k] = yv[k - 1]; dyv[k] = dyv[k - 1]; }
            if (j0 < N) { float2 p = ydy[j0]; yv[0] = p.x; dyv[0] = p.y; }

            float upn = buf[(d + 1) & 1][tm1];   // neighbor bottom row, 1 step back

            #pragma unroll
            for (int k = R - 1; k >= 0; --k) {   // descending: cur[k-1] still prev-step
                const int jk = j0 - k;
                if (jk >= 0 && jk < N) {
                    float m    = xv[k] - yv[k];
                    float mm   = m * m;
                    float up_c = (k == 0) ? upn : cur[k - 1];
                    float dg_c = (k == 0) ? dgN : oldv[k - 1];
                    const bool hasu = (k > 0) || (t > 0);
                    float d1 = hasu ? ((jk > 0) ? dg_c + mm : BIGF)
                                    : ((jk == 0) ? mm : BIGF);       // c(0,0)=(x0-y0)^2
                    float d2 = hasu ? up_c + transf2(dxv[k], dx2[k], m) : BIGF;
                    float d3 = (jk > 0) ? cur[k] + transf(dyv[k], -m) : BIGF;
                    float c  = softmin3f(d1, d2, d3);
                    oldv[k] = cur[k];            // becomes next step's diag for row k+1
                    cur[k]  = c;
                }
            }
            dgN = upn;                           // becomes row 0's diag next step

            const int jb = j0 - (R - 1);         // bottom-row publish
            if (jb >= 0 && jb < N) buf[d & 1][t] = cur[R - 1];
        }
        __syncthreads();
    }

    if (t == T - 1) ws[b] = cur[R - 1];          // c(N-1, N-1)
}

__global__ void reduce_mean_kernel(const float* __restrict__ ws,
                                   float* __restrict__ out, int B) {
    if (threadIdx.x == 0 && blockIdx.x == 0) {
        float s = 0.0f;
        for (int i = 0; i < B; ++i) s += ws[i];
        out[0] = s / (float)B;
    }
}

extern "C" void kernel_launch(void* const* d_in, const int* in_sizes, int n_in,
                              void* d_out, int out_size, void* d_ws, size_t ws_size,
                              hipStream_t stream) {
    const float* x = (const float*)d_in[0];   // (32, 1, 1024) f32
    const float* y = (const float*)d_in[1];   // (32, 1, 1024) f32
    const int B = in_sizes[0] / N;            // 32
    float* ws = (float*)d_ws;

    soft_msm_kernel<<<B, T, 0, stream>>>(x, y, ws);
    reduce_mean_kernel<<<1, 32, 0, stream>>>(ws, (float*)d_out, B);
}